// DREAMReconstructor_71270687310083
// MI455X (gfx1250) — compile-verified
//
#include <hip/hip_runtime.h>
#include <math.h>

// DREAM reconstructor: recurrent scan h_t = tanh(W_in x_t + (h B_low^T) A^T + b),
// decoder recon_t = W_dec h_t + b_dec.  Fully fused persistent kernel:
// 16 workgroups x 16 waves (wave32), each WG owns a 16-row batch tile and loops t.
// All matmuls via V_WMMA_F32_16X16X4_F32; all weights resident in LDS.

typedef float v2f __attribute__((ext_vector_type(2)));
typedef float v8f __attribute__((ext_vector_type(8)));

#define BATCH 256
#define TIME  1024
#define IND   39
#define HID   256
#define RANK  16

// LDS strides (in floats), padded for bank spreading (64 banks)
#define WIS 264   // W_in^T  [40][WIS]
#define BLS 16    // B_low^T [256][BLS]
#define AWS 264   // A^T     [16][AWS]
#define WDS 48    // W_dec^T [256][WDS]
#define XS  44    // x tile  [16][XS]
#define HS  260   // h tile  [16][HS]
#define RS  20    // r       [16][RS]

// LDS layout offsets (floats)
#define OFF_WI 0
#define OFF_BL (OFF_WI + 40*WIS)
#define OFF_AW (OFF_BL + 256*BLS)
#define OFF_WD (OFF_AW + 16*AWS)
#define OFF_B  (OFF_WD + 256*WDS)
#define OFF_BD (OFF_B  + 256)
#define OFF_X  (OFF_BD + 48)
#define OFF_H  (OFF_X  + 16*XS)
#define OFF_R  (OFF_H  + 16*HS)
#define OFF_RP (OFF_R  + 16*RS)
#define OFF_DP (OFF_RP + 16*256)
#define SMEM_FLOATS (OFF_DP + 16*768)
#define SMEM_BYTES  (SMEM_FLOATS * 4)     // ~212 KB < 320 KB/WGP

// One f32 WMMA step: D(16x16) = A(16x4) * B(4x16) + C
__device__ __forceinline__ v8f wmma4(v2f a, v2f b, v8f c) {
  return __builtin_amdgcn_wmma_f32_16x16x4_f32(
      /*neg_a=*/false, a, /*neg_b=*/false, b,
      /*c_mod=*/(short)0, c, /*reuse_a=*/false, /*reuse_b=*/false);
}

// Hardware V_TANH_F32 when available (gfx1250 TRANS op), else ocml tanh.
__device__ __forceinline__ float fast_tanh(float v) {
#if __has_builtin(__builtin_amdgcn_tanhf)
  return __builtin_amdgcn_tanhf(v);
#else
  return tanhf(v);
#endif
}

// A-fragment: element[v] = src[m][kk + 2*(lane>=16) + v]   (row-major, stride)
__device__ __forceinline__ v2f ldA(const float* base, int stride, int m, int kk, int kh) {
  const float* p = base + m * stride + kk + 2 * kh;
  v2f a; a.x = p[0]; a.y = p[1]; return a;
}
// B-fragment: element[v] = src[kk + 2*(lane>=16) + v][col]  (row-major, stride)
__device__ __forceinline__ v2f ldB(const float* base, int stride, int col, int kk, int kh) {
  const float* p = base + (kk + 2 * kh) * stride + col;
  v2f b; b.x = p[0]; b.y = p[stride]; return b;
}

__global__ __launch_bounds__(512)
void dream_scan(const float* __restrict__ x, const float* __restrict__ W_in,
                const float* __restrict__ Amat, const float* __restrict__ B_low,
                const float* __restrict__ bvec, const float* __restrict__ W_dec,
                const float* __restrict__ b_dec, float* __restrict__ out) {
  extern __shared__ float sm[];
  float* Wi = sm + OFF_WI;
  float* Bl = sm + OFF_BL;
  float* Aw = sm + OFF_AW;
  float* Wd = sm + OFF_WD;
  float* Bs = sm + OFF_B;
  float* Bd = sm + OFF_BD;
  float* Xl = sm + OFF_X;
  float* Hl = sm + OFF_H;
  float* Rl = sm + OFF_R;
  float* Rp = sm + OFF_RP;
  float* Dp = sm + OFF_DP;

  const int tid  = threadIdx.x;
  const int wave = tid >> 5;
  const int lane = tid & 31;
  const int lm   = lane & 15;    // M-row (A/C) or N-col (B/C) within tile
  const int kh   = lane >> 4;    // K-half select
  const int b0   = blockIdx.x * 16;   // batch tile base
  const int n0   = wave * 16;         // hidden-column tile base

  // ---- prologue: stage (transposed) weights into LDS, zero h and x pads ----
  for (int e = tid; e < 40 * 256; e += 512) {
    int n = e & 255, k = e >> 8;
    Wi[k * WIS + n] = (k < IND) ? W_in[n * IND + k] : 0.0f;
  }
  for (int e = tid; e < 256 * 16; e += 512) {
    int j = e & 15, k = e >> 4;
    Bl[k * BLS + j] = B_low[j * HID + k];
  }
  for (int e = tid; e < 16 * 256; e += 512) {
    int n = e & 255, j = e >> 8;
    Aw[j * AWS + n] = Amat[n * RANK + j];
  }
  for (int e = tid; e < 256 * 48; e += 512) {
    int c = e % 48, k = e / 48;
    Wd[k * WDS + c] = (c < IND) ? W_dec[c * HID + k] : 0.0f;
  }
  if (tid < 256) Bs[tid] = bvec[tid];
  if (tid < 48)  Bd[tid] = (tid < IND) ? b_dec[tid] : 0.0f;
  for (int e = tid; e < 16 * HS; e += 512) Hl[e] = 0.0f;  // h0 = 0
  for (int e = tid; e < 16 * XS; e += 512) Xl[e] = 0.0f;  // zero K-pad cols
  __syncthreads();

  // ---- hoisted loop-invariant descriptors ----
  // x prefetch: thread covers elements e0=tid and e1=tid+512 of the 16x39 tile
  const int  e0 = tid, e1 = tid + 512;
  const bool xv0 = (e0 < 16 * IND), xv1 = (e1 < 16 * IND);
  const float* xg0 = x;  int xl0 = 0;
  const float* xg1 = x;  int xl1 = 0;
  if (xv0) { int r = e0 / IND, c = e0 % IND;
    xg0 = x + (size_t)(b0 + r) * TIME * IND + c;  xl0 = r * XS + c; }
  if (xv1) { int r = e1 / IND, c = e1 % IND;
    xg1 = x + (size_t)(b0 + r) * TIME * IND + c;  xl1 = r * XS + c; }

  // recon store: thread covers elements ea=tid and eb=tid+512 of the 16x48 tile
  const int  ea = tid, eb = tid + 512;
  const bool sva = ((ea % 48) < IND);
  const bool svb = (eb < 768) && ((eb % 48) < IND);
  float* outa = out;  float bda = 0.0f;
  float* outb = out;  float bdb = 0.0f;
  if (sva) { int row = ea / 48, c = ea % 48;
    outa = out + (size_t)(b0 + row) * TIME * IND + c;  bda = Bd[c]; }
  if (svb) { int row = eb / 48, c = eb % 48;
    outb = out + (size_t)(b0 + row) * TIME * IND + c;  bdb = Bd[c]; }
  const float bb = Bs[n0 + lm];

  // preload x_0 into registers
  float xr0 = 0.0f, xr1 = 0.0f;
  if (xv0) xr0 = xg0[0];
  if (xv1) xr1 = xg1[0];

  v8f h_acc = {};

  for (int t = 0; t < TIME; ++t) {
    // stage x_t into LDS
    if (xv0) Xl[xl0] = xr0;
    if (xv1) Xl[xl1] = xr1;
    __syncthreads();                                   // B1
    // prefetch x_{t+1} into registers (latency overlaps whole step)
    if (t + 1 < TIME) {
      size_t off = (size_t)(t + 1) * IND;
      if (xv0) xr0 = xg0[off];
      if (xv1) xr1 = xg1[off];
    }

    // pre = b + x @ W_in^T   (K padded to 40 -> 10 WMMAs)
    v8f acc;
    #pragma unroll
    for (int v = 0; v < 8; ++v) acc[v] = bb;
    #pragma unroll
    for (int kk = 0; kk < 40; kk += 4)
      acc = wmma4(ldA(Xl, XS, lm, kk, kh), ldB(Wi, WIS, n0 + lm, kk, kh), acc);

    // r = h @ B_low^T : split-K across waves (4 WMMAs each), reduce in LDS
    {
      v8f rp = {};
      const int kb = wave * 16;
      #pragma unroll
      for (int kk = 0; kk < 16; kk += 4)
        rp = wmma4(ldA(Hl, HS, lm, kb + kk, kh), ldB(Bl, BLS, lm, kb + kk, kh), rp);
      #pragma unroll
      for (int v = 0; v < 8; ++v)
        Rp[wave * 256 + (v + 8 * kh) * 16 + lm] = rp[v];
    }
    __syncthreads();                                   // B2
    if (tid < 256) {
      float s = 0.0f;
      #pragma unroll
      for (int w = 0; w < 16; ++w) s += Rp[w * 256 + tid];
      Rl[(tid >> 4) * RS + (tid & 15)] = s;
    }
    __syncthreads();                                   // B3

    // pre += r @ A^T  (K=16 -> 4 WMMAs), then tanh -> h_t
    #pragma unroll
    for (int kk = 0; kk < 16; kk += 4)
      acc = wmma4(ldA(Rl, RS, lm, kk, kh), ldB(Aw, AWS, n0 + lm, kk, kh), acc);
    #pragma unroll
    for (int v = 0; v < 8; ++v) acc[v] = fast_tanh(acc[v]);
    h_acc = acc;
    #pragma unroll
    for (int v = 0; v < 8; ++v) Hl[(v + 8 * kh) * HS + n0 + lm] = acc[v];
    __syncthreads();                                   // B4

    // recon_t = h @ W_dec^T + b_dec : split-K (12 WMMAs/wave), reduce, store
    {
      v8f d0 = {}, d1 = {}, d2 = {};
      const int kb = wave * 16;
      #pragma unroll
      for (int kk = 0; kk < 16; kk += 4) {
        v2f a = ldA(Hl, HS, lm, kb + kk, kh);
        d0 = wmma4(a, ldB(Wd, WDS,  0 + lm, kb + kk, kh), d0);
        d1 = wmma4(a, ldB(Wd, WDS, 16 + lm, kb + kk, kh), d1);
        d2 = wmma4(a, ldB(Wd, WDS, 32 + lm, kb + kk, kh), d2);
      }
      #pragma unroll
      for (int v = 0; v < 8; ++v) {
        int row = v + 8 * kh;
        Dp[wave * 768 + row * 48 +  0 + lm] = d0[v];
        Dp[wave * 768 + row * 48 + 16 + lm] = d1[v];
        Dp[wave * 768 + row * 48 + 32 + lm] = d2[v];
      }
    }
    __syncthreads();                                   // B5
    {
      const size_t toff = (size_t)t * IND;
      if (sva) {
        float s = bda;
        #pragma unroll
        for (int w = 0; w < 16; ++w) s += Dp[w * 768 + ea];
        outa[toff] = s;
      }
      if (svb) {
        float s = bdb;
        #pragma unroll
        for (int w = 0; w < 16; ++w) s += Dp[w * 768 + eb];
        outb[toff] = s;
      }
    }
    // next-iter LDS overwrites are all separated from this iter's reads by >=1 barrier
  }

  // h_final -> d_out tail (after the B*T*D recon block)
  {
    float* hf = out + (size_t)BATCH * TIME * IND;
    #pragma unroll
    for (int v = 0; v < 8; ++v)
      hf[(size_t)(b0 + v + 8 * kh) * HID + n0 + lm] = h_acc[v];
  }
}

extern "C" void kernel_launch(void* const* d_in, const int* in_sizes, int n_in,
                              void* d_out, int out_size, void* d_ws, size_t ws_size,
                              hipStream_t stream) {
  const float* x     = (const float*)d_in[0];
  const float* W_in  = (const float*)d_in[1];
  const float* Amat  = (const float*)d_in[2];
  const float* B_low = (const float*)d_in[3];
  const float* bvec  = (const float*)d_in[4];
  const float* W_dec = (const float*)d_in[5];
  const float* b_dec = (const float*)d_in[6];
  float* out = (float*)d_out;

  hipFuncSetAttribute((const void*)dream_scan,
                      hipFuncAttributeMaxDynamicSharedMemorySize, SMEM_BYTES);
  dream_scan<<<BATCH / 16, 512, SMEM_BYTES, stream>>>(
      x, W_in, Amat, B_low, bvec, W_dec, b_dec, out);
}